// Decoder_10488310136894
// MI455X (gfx1250) — compile-verified
//
#include <hip/hip_runtime.h>
#include <math.h>

// ---------------------------------------------------------------------------
// LSTM teacher-forced decoder for MI455X (gfx1250, wave32).
//   V=50257 vocab, I=H=1024, L=50 steps.
// Strategy:
//   (1) gather embeddings + init h/c + zero pads
//   (2) precomp[64,4096] = inputs_pad @ W_ih^T + b_ih + b_hh   (f32 WMMA GEMM)
//   (3) 50x { gates = precomp[t] + W_hh @ h  (wave-per-row matvec);
//             cell: h,c update + write hs_pad[t] }
//   (4) logits[50,50257] = hs_pad @ W_out^T + b_out            (f32 WMMA GEMM,
//       memory-bound: streams 206MB of W_out once, ~9us at 23.3 TB/s)
// ---------------------------------------------------------------------------

typedef __attribute__((ext_vector_type(2))) float v2f;
typedef __attribute__((ext_vector_type(8))) float v8f;

#define VOCAB 50257
#define HID   1024
#define INP   1024
#define SEQL  50
#define G4H   4096
#define START_ID 1
#define MPAD  64          // L=50 padded to 4 tiles of 16 rows

// ---- (1) gather + init -----------------------------------------------------
__global__ void gather_init_kernel(const int* __restrict__ tokens,
                                   const float* __restrict__ embedding,
                                   const float* __restrict__ h0,
                                   const float* __restrict__ c0,
                                   float* __restrict__ inputs_pad, // [64,1024]
                                   float* __restrict__ hs_pad,     // [64,1024]
                                   float* __restrict__ hbuf,       // [1024]
                                   float* __restrict__ cbuf)       // [1024]
{
    int b = blockIdx.x;
    int t = threadIdx.x;
    if (b < MPAD) {
        // row 0 = <START> embedding; rows 1..49 = emb(tokens[r-1]); 50..63 = 0
        const float* src = nullptr;
        if (b == 0)          src = embedding + (size_t)START_ID * INP;
        else if (b < SEQL)   src = embedding + (size_t)tokens[b - 1] * INP;
        float* dst = inputs_pad + (size_t)b * INP;
        for (int k = t; k < INP; k += blockDim.x)
            dst[k] = src ? src[k] : 0.0f;
    } else if (b == MPAD) {
        for (int k = t; k < HID; k += blockDim.x) {
            hbuf[k] = h0[k];
            cbuf[k] = c0[k];
        }
    } else {
        // zero padded hs rows 50..63 (harness poisons ws with 0xAA)
        float* dst = hs_pad + (size_t)SEQL * HID;
        int n = (MPAD - SEQL) * HID;
        for (int k = t; k < n; k += blockDim.x) dst[k] = 0.0f;
    }
}

// ---- (2)/(4) f32 WMMA GEMM:  C[m,n] = sum_k A[m,k]*W[n,k] + bias[n] (+bias2)
// A: [MPAD, K] row-major (rows >= Mstore are zero), K multiple of 4.
// W: [Nw, K] row-major; load row clamped, store guarded (n < Nstore, m < Mstore).
// One 16x16 output tile per wave; 4 M-tiles per N-tile are adjacent waves so
// the tile's 16 W rows are reused from cache by 4 waves in the same block.
__global__ void wmma_gemm_kernel(const float* __restrict__ A,
                                 const float* __restrict__ W,
                                 const float* __restrict__ bias,
                                 const float* __restrict__ bias2,
                                 float* __restrict__ C,
                                 int K, int Nw, int Ntiles,
                                 int Mstore, int Nstore, int ldC)
{
    int waveId = (int)((blockIdx.x * blockDim.x + threadIdx.x) >> 5);
    int totalTiles = Ntiles * (MPAD / 16);
    if (waveId >= totalTiles) return;

    int mtile = waveId & 3;          // MPAD/16 == 4
    int ntile = waveId >> 2;
    int m0 = mtile * 16;
    int n0 = ntile * 16;

    int lane = threadIdx.x & 31;
    int l15  = lane & 15;
    int koff = (lane >> 4) << 1;     // lanes 0-15: K+{0,1}; lanes 16-31: K+{2,3}

    const float* Arow = A + (size_t)(m0 + l15) * K + koff;
    int ncol = n0 + l15;
    int nld  = (ncol < Nw) ? ncol : (Nw - 1);   // clamp OOB column loads
    const float* Wrow = W + (size_t)nld * K + koff;

    v8f acc = {};
#pragma unroll 4
    for (int k0 = 0; k0 < K; k0 += 4) {
        v2f a = *(const v2f*)(Arow + k0);   // {A[m][k], A[m][k+1]} per ISA layout
        v2f b = *(const v2f*)(Wrow + k0);   // {W[n][k], W[n][k+1]} == B^T cols
        acc = __builtin_amdgcn_wmma_f32_16x16x4_f32(
            /*neg_a=*/false, a, /*neg_b=*/false, b,
            /*c_mod=*/(short)0, acc, /*reuse_a=*/false, /*reuse_b=*/false);
    }

    float bs = 0.0f;
    if (ncol < Nstore) {
        bs = bias[ncol];
        if (bias2) bs += bias2[ncol];
    }
    int mbase = m0 + ((lane >> 4) << 3);     // lanes 16-31 hold rows M+8
#pragma unroll
    for (int r = 0; r < 8; ++r) {
        int m = mbase + r;
        if (m < Mstore && ncol < Nstore)
            C[(size_t)m * ldC + ncol] = acc[r] + bs;
    }
}

// ---- (3a) gates[j] = precomp_t[j] + dot(W_hh[j], h) ; one wave per row -----
__global__ void lstm_matvec_kernel(const float* __restrict__ Whh, // [4096,1024]
                                   const float* __restrict__ h,   // [1024]
                                   const float* __restrict__ pre_t,
                                   float* __restrict__ gates)     // [4096]
{
    int wave = (int)((blockIdx.x * blockDim.x + threadIdx.x) >> 5);
    int lane = threadIdx.x & 31;
    if (wave >= G4H) return;

    const float* wrow = Whh + (size_t)wave * HID;
    float sum = 0.0f;
#pragma unroll
    for (int i = 0; i < 8; ++i) {
        int k = i * 128 + lane * 4;          // coalesced 512B per wave per iter
        float4 w4 = *(const float4*)(wrow + k);
        float4 h4 = *(const float4*)(h + k);
        sum = fmaf(w4.x, h4.x, sum);
        sum = fmaf(w4.y, h4.y, sum);
        sum = fmaf(w4.z, h4.z, sum);
        sum = fmaf(w4.w, h4.w, sum);
    }
#pragma unroll
    for (int off = 16; off > 0; off >>= 1)
        sum += __shfl_down(sum, off, 32);    // wave32 reduction
    if (lane == 0) gates[wave] = sum + pre_t[wave];
}

// ---- (3b) LSTM cell elementwise update ------------------------------------
__global__ void lstm_cell_kernel(const float* __restrict__ gates, // [4096] i,f,g,o
                                 float* __restrict__ h,
                                 float* __restrict__ c,
                                 float* __restrict__ hs_t)        // hs_pad + t*H
{
    int j = blockIdx.x * blockDim.x + threadIdx.x;
    if (j >= HID) return;
    float ig = 1.0f / (1.0f + expf(-gates[j]));
    float fg = 1.0f / (1.0f + expf(-gates[HID + j]));
    float gg = tanhf(gates[2 * HID + j]);
    float og = 1.0f / (1.0f + expf(-gates[3 * HID + j]));
    float cn = fg * c[j] + ig * gg;
    float hn = og * tanhf(cn);
    c[j] = cn;
    h[j] = hn;
    hs_t[j] = hn;
}

// ---------------------------------------------------------------------------
extern "C" void kernel_launch(void* const* d_in, const int* in_sizes, int n_in,
                              void* d_out, int out_size, void* d_ws, size_t ws_size,
                              hipStream_t stream)
{
    const int*   tokens = (const int*)  d_in[0];
    const float* h0     = (const float*)d_in[1];
    const float* c0     = (const float*)d_in[2];
    const float* emb    = (const float*)d_in[3];
    const float* W_ih   = (const float*)d_in[4];
    const float* W_hh   = (const float*)d_in[5];
    const float* b_ih   = (const float*)d_in[6];
    const float* b_hh   = (const float*)d_in[7];
    const float* W_out  = (const float*)d_in[8];
    const float* b_out  = (const float*)d_in[9];
    float* logits = (float*)d_out;

    // workspace layout (floats): ~1.6 MB total
    float* ws         = (float*)d_ws;
    float* inputs_pad = ws;                         // 64*1024
    float* precomp    = inputs_pad + MPAD * INP;    // 64*4096
    float* hs_pad     = precomp + MPAD * G4H;       // 64*1024
    float* hbuf       = hs_pad + MPAD * HID;        // 1024
    float* cbuf       = hbuf + HID;                 // 1024
    float* gates      = cbuf + HID;                 // 4096

    // (1) gather embeddings, init h/c, zero pad rows
    gather_init_kernel<<<MPAD + 2, 256, 0, stream>>>(
        tokens, emb, h0, c0, inputs_pad, hs_pad, hbuf, cbuf);

    // (2) precomp = inputs_pad @ W_ih^T + (b_ih + b_hh)
    {
        int Ntiles = G4H / 16;                      // 256
        int tiles  = Ntiles * (MPAD / 16);          // 1024 waves
        int blocks = (tiles * 32 + 255) / 256;
        wmma_gemm_kernel<<<blocks, 256, 0, stream>>>(
            inputs_pad, W_ih, b_ih, b_hh, precomp,
            INP, G4H, Ntiles, MPAD, G4H, G4H);
    }

    // (3) 50 sequential LSTM steps (stream ordering carries the dependency)
    for (int t = 0; t < SEQL; ++t) {
        lstm_matvec_kernel<<<(G4H * 32) / 256, 256, 0, stream>>>(
            W_hh, hbuf, precomp + (size_t)t * G4H, gates);
        lstm_cell_kernel<<<HID / 256, 256, 0, stream>>>(
            gates, hbuf, cbuf, hs_pad + (size_t)t * HID);
    }

    // (4) logits = hs_pad @ W_out^T + b_out   (memory-bound: 206MB of W_out)
    {
        int Ntiles = (VOCAB + 15) / 16;             // 3142
        int tiles  = Ntiles * (MPAD / 16);          // 12568 waves
        int blocks = (tiles * 32 + 255) / 256;
        wmma_gemm_kernel<<<blocks, 256, 0, stream>>>(
            hs_pad, W_out, b_out, nullptr, logits,
            HID, VOCAB, Ntiles, SEQL, VOCAB, VOCAB);
    }
}